// LargeGCN_62895501082703
// MI455X (gfx1250) — compile-verified
//
#include <hip/hip_runtime.h>
#include <hip/hip_bf16.h>

typedef __attribute__((ext_vector_type(2))) float v2f;
typedef __attribute__((ext_vector_type(8))) float v8f;

#define FDIM 64
#define BN_EPS 1e-5f

// ---------------------------------------------------------------------------
// Degree / normalization precompute
// ---------------------------------------------------------------------------
__global__ __launch_bounds__(256)
void gcn_init_deg(float* __restrict__ deg, int N) {
    int i = blockIdx.x * blockDim.x + threadIdx.x;
    if (i < N) deg[i] = 1.0f;               // self-loop
}

__global__ __launch_bounds__(256)
void gcn_deg_accum(const int* __restrict__ dst, float* __restrict__ deg, int E) {
    int e = blockIdx.x * blockDim.x + threadIdx.x;
    if (e < E) atomicAdd(&deg[dst[e]], 1.0f);
}

__global__ __launch_bounds__(256)
void gcn_deg_finalize(const float* __restrict__ deg, float* __restrict__ dis,
                      float* __restrict__ selfn, int N) {
    int i = blockIdx.x * blockDim.x + threadIdx.x;
    if (i < N) {
        float d = deg[i];
        dis[i]   = rsqrtf(d);               // D^{-1/2}
        selfn[i] = 1.0f / d;                // self-loop norm
    }
}

// ---------------------------------------------------------------------------
// Fused GEMM: hw = hin @ W  (f32 WMMA 16x16x4, K=64 => 16 WMMA steps)
// also seeds acc = selfn * hw + bias (self-loop term + bias)
// One block = 16 rows; 4 waves cover the 64 output columns (16 each).
// ---------------------------------------------------------------------------
__global__ __launch_bounds__(128)
void gcn_gemm_wmma(const float* __restrict__ hin, const float* __restrict__ W,
                   const float* __restrict__ bias, const float* __restrict__ selfn,
                   float* __restrict__ hw, float* __restrict__ acc) {
    const int lane    = threadIdx.x & 31;
    const int wave    = threadIdx.x >> 5;        // 0..3 -> column tile
    const int l15     = lane & 15;
    const int hsel    = lane >> 4;               // which K-half this lane holds
    const int rowBase = blockIdx.x << 4;
    const int row     = rowBase + l15;           // A-matrix row (M)
    const int col     = (wave << 4) + l15;       // B/C/D column (N)

    v8f c = {};                                  // f32 16x16 accumulator (8 VGPRs)
    #pragma unroll
    for (int k0 = 0; k0 < FDIM; k0 += 4) {
        const int ka = k0 + 2 * hsel;
        v2f a, b;
        // A 16x4: lanes 0-15 hold K={k0,k0+1}, lanes 16-31 hold K={k0+2,k0+3}
        a.x = hin[(size_t)row * FDIM + ka];
        a.y = hin[(size_t)row * FDIM + ka + 1];
        // B 4x16: rows striped across lanes, same K split across lane halves
        b.x = W[(size_t)ka * FDIM + col];
        b.y = W[(size_t)(ka + 1) * FDIM + col];
        c = __builtin_amdgcn_wmma_f32_16x16x4_f32(
                /*neg_a=*/false, a, /*neg_b=*/false, b,
                /*c_mod=*/(short)0, c, /*reuse_a=*/false, /*reuse_b=*/false);
    }

    const float bcol = bias[col];
    #pragma unroll
    for (int v = 0; v < 8; ++v) {
        const int m = rowBase + v + 8 * hsel;    // D layout: VGPR v -> M=v (+8 for hi lanes)
        const float val = c[v];
        hw [(size_t)m * FDIM + col] = val;
        acc[(size_t)m * FDIM + col] = fmaf(selfn[m], val, bcol);
    }
}

// ---------------------------------------------------------------------------
// Edge scatter: acc[dst] += dis[src]*dis[dst] * hw[src]
// One wave per edge: 32 lanes x float2 = 64 features, coalesced b64 accesses.
// ---------------------------------------------------------------------------
__global__ __launch_bounds__(256)
void gcn_scatter(const float* __restrict__ hw, float* __restrict__ acc,
                 const float* __restrict__ dis, const int* __restrict__ src,
                 const int* __restrict__ dst, int E) {
    const int lane = threadIdx.x & 31;
    const int w    = (blockIdx.x * blockDim.x + threadIdx.x) >> 5;
    const int nw   = (gridDim.x * blockDim.x) >> 5;
    for (int e = w; e < E; e += nw) {
        const int s = src[e];
        const int d = dst[e];
        const float en = dis[s] * dis[d];
        const float2 v = *(const float2*)(hw + (size_t)s * FDIM + 2 * lane);
        atomicAdd(acc + (size_t)d * FDIM + 2 * lane,     en * v.x);
        atomicAdd(acc + (size_t)d * FDIM + 2 * lane + 1, en * v.y);
    }
}

// ---------------------------------------------------------------------------
// BatchNorm: two-stage column sum / sum-of-squares reduction
// ---------------------------------------------------------------------------
__global__ __launch_bounds__(128)
void gcn_zero_bn(float* __restrict__ bsum, float* __restrict__ bsq) {
    if (threadIdx.x < FDIM) { bsum[threadIdx.x] = 0.0f; bsq[threadIdx.x] = 0.0f; }
}

__global__ __launch_bounds__(256)
void gcn_bn_reduce(const float* __restrict__ acc, float* __restrict__ bsum,
                   float* __restrict__ bsq, int N) {
    const int col = threadIdx.x & 63;
    const int rg  = threadIdx.x >> 6;            // 4 row-threads per column
    float s = 0.0f, q = 0.0f;
    for (int n = blockIdx.x * 4 + rg; n < N; n += gridDim.x * 4) {
        const float v = acc[(size_t)n * FDIM + col];
        s += v; q += v * v;
    }
    __shared__ float ssum[256];
    __shared__ float ssq[256];
    ssum[threadIdx.x] = s; ssq[threadIdx.x] = q;
    __syncthreads();
    if (threadIdx.x < FDIM) {
        s = ssum[threadIdx.x] + ssum[threadIdx.x + 64] +
            ssum[threadIdx.x + 128] + ssum[threadIdx.x + 192];
        q = ssq[threadIdx.x] + ssq[threadIdx.x + 64] +
            ssq[threadIdx.x + 128] + ssq[threadIdx.x + 192];
        atomicAdd(&bsum[threadIdx.x], s);
        atomicAdd(&bsq[threadIdx.x], q);
    }
}

__global__ __launch_bounds__(64)
void gcn_bn_finalize(const float* __restrict__ bsum, const float* __restrict__ bsq,
                     const float* __restrict__ gamma, const float* __restrict__ beta,
                     float* __restrict__ scale, float* __restrict__ shift, int N) {
    const int c = threadIdx.x;
    if (c < FDIM) {
        const float inv = 1.0f / (float)N;
        const float m   = bsum[c] * inv;
        const float var = bsq[c] * inv - m * m;     // biased variance
        const float sc  = gamma[c] * rsqrtf(var + BN_EPS);
        scale[c] = sc;
        shift[c] = beta[c] - m * sc;
    }
}

__global__ __launch_bounds__(256)
void gcn_bn_relu(const float* __restrict__ acc, const float* __restrict__ scale,
                 const float* __restrict__ shift, float* __restrict__ h, long total) {
    const long i = (long)blockIdx.x * blockDim.x + threadIdx.x;
    if (i < total) {
        const int c = (int)(i & 63);
        h[i] = fmaxf(0.0f, fmaf(acc[i], scale[c], shift[c]));
    }
}

// ---------------------------------------------------------------------------
// Final FC (64 -> 1): one wave per node, shuffle reduction
// ---------------------------------------------------------------------------
__global__ __launch_bounds__(256)
void gcn_fc(const float* __restrict__ acc, const float* __restrict__ fcw,
            const float* __restrict__ fcb, float* __restrict__ out, int N) {
    const int lane = threadIdx.x & 31;
    const int n    = (blockIdx.x * blockDim.x + threadIdx.x) >> 5;
    if (n >= N) return;
    const float2 v = *(const float2*)(acc + (size_t)n * FDIM + 2 * lane);
    const float2 w = *(const float2*)(fcw + 2 * lane);
    float p = v.x * w.x + v.y * w.y;
    #pragma unroll
    for (int off = 16; off > 0; off >>= 1) p += __shfl_xor(p, off, 32);
    if (lane == 0) out[n] = p + fcb[0];
}

// ---------------------------------------------------------------------------
// Launcher
// ---------------------------------------------------------------------------
extern "C" void kernel_launch(void* const* d_in, const int* in_sizes, int n_in,
                              void* d_out, int out_size, void* d_ws, size_t ws_size,
                              hipStream_t stream) {
    const float* x   = (const float*)d_in[0];
    const int*   ei  = (const int*)  d_in[1];   // [2, E]: src = ei[0:E), dst = ei[E:2E)
    const float* W0  = (const float*)d_in[2];
    const float* b0  = (const float*)d_in[3];
    const float* g0  = (const float*)d_in[4];
    const float* bt0 = (const float*)d_in[5];
    const float* W1  = (const float*)d_in[6];
    const float* b1  = (const float*)d_in[7];
    const float* g1  = (const float*)d_in[8];
    const float* bt1 = (const float*)d_in[9];
    const float* W2  = (const float*)d_in[10];
    const float* b2  = (const float*)d_in[11];
    const float* fcw = (const float*)d_in[12];
    const float* fcb = (const float*)d_in[13];

    const int N = in_sizes[0] / FDIM;
    const int E = in_sizes[1] / 2;
    const int* src = ei;
    const int* dst = ei + E;

    // workspace carve-up (all f32)
    float* deg   = (float*)d_ws;
    float* dis   = deg   + N;
    float* selfn = dis   + N;
    float* bsum  = selfn + N;
    float* bsq   = bsum  + FDIM;
    float* scale = bsq   + FDIM;
    float* shift = scale + FDIM;
    float* h     = shift + FDIM;               // N*64
    float* hw    = h     + (size_t)N * FDIM;   // N*64
    float* acc   = hw    + (size_t)N * FDIM;   // N*64

    // degree / normalization
    gcn_init_deg    <<<(N + 255) / 256, 256, 0, stream>>>(deg, N);
    gcn_deg_accum   <<<(E + 255) / 256, 256, 0, stream>>>(dst, deg, E);
    gcn_deg_finalize<<<(N + 255) / 256, 256, 0, stream>>>(deg, dis, selfn, N);

    const float* Ws[3]  = { W0, W1, W2 };
    const float* bs[3]  = { b0, b1, b2 };
    const float* gs[2]  = { g0, g1 };
    const float* bts[2] = { bt0, bt1 };

    const int gemmBlocks = N >> 4;             // N divisible by 16 (N=100000)
    const long total = (long)N * FDIM;
    const float* hin = x;

    for (int l = 0; l < 3; ++l) {
        gcn_gemm_wmma<<<gemmBlocks, 128, 0, stream>>>(hin, Ws[l], bs[l], selfn, hw, acc);
        gcn_scatter  <<<4096, 256, 0, stream>>>(hw, acc, dis, src, dst, E);
        if (l < 2) {
            gcn_zero_bn    <<<1, 128, 0, stream>>>(bsum, bsq);
            gcn_bn_reduce  <<<256, 256, 0, stream>>>(acc, bsum, bsq, N);
            gcn_bn_finalize<<<1, 64, 0, stream>>>(bsum, bsq, gs[l], bts[l], scale, shift, N);
            gcn_bn_relu    <<<(int)((total + 255) / 256), 256, 0, stream>>>(acc, scale, shift, h, total);
            hin = h;
        }
    }

    gcn_fc<<<(N * 32 + 255) / 256, 256, 0, stream>>>(acc, fcw, fcb, (float*)d_out, N);
}